// WaveEncoder_15418932592841
// MI455X (gfx1250) — compile-verified
//
#include <hip/hip_runtime.h>

typedef __attribute__((ext_vector_type(2))) float v2f;
typedef __attribute__((ext_vector_type(8))) float v8f;

#define B_  64
#define T_  4096
#define F_  8
#define RC_ 32
#define NL_ 23
#define TN_ 64   // time columns per block

__device__ __forceinline__ float fsigmoid(float x) { return 1.0f / (1.0f + __expf(-x)); }
__device__ __forceinline__ float ftanh_(float x)   { return 1.0f - 2.0f / (__expf(2.0f * x) + 1.0f); }

// ---------------------------------------------------------------------------
// Kernel 0: h0 = tanh(W_h @ [x;features] + b_h), c0 = tanh(W_c @ ... + b_c)
// one thread per (b, t); 32x9 matvec each. Trivial compute, coalesced stores.
// ---------------------------------------------------------------------------
__global__ __launch_bounds__(256)
void wave_init_kernel(const float* __restrict__ x,
                      const float* __restrict__ feat,
                      const float* __restrict__ w_h,
                      const float* __restrict__ b_h,
                      const float* __restrict__ w_c,
                      const float* __restrict__ b_c,
                      float* __restrict__ h0,
                      float* __restrict__ c0) {
  int idx = blockIdx.x * blockDim.x + threadIdx.x;   // (b, t)
  int b = idx >> 12;           // / T_
  int t = idx & (T_ - 1);
  float in[9];
  in[0] = x[(size_t)b * T_ + t];
#pragma unroll
  for (int f = 0; f < F_; ++f)
    in[1 + f] = feat[((size_t)b * F_ + f) * T_ + t];
  for (int o = 0; o < RC_; ++o) {
    float ah = b_h[o], ac = b_c[o];
#pragma unroll
    for (int cc = 0; cc < 9; ++cc) {
      ah = fmaf(w_h[o * 9 + cc], in[cc], ah);
      ac = fmaf(w_c[o * 9 + cc], in[cc], ac);
    }
    size_t oidx = ((size_t)b * RC_ + o) * T_ + t;
    h0[oidx] = ftanh_(ah);
    c0[oidx] = ftanh_(ac);
  }
}

// ---------------------------------------------------------------------------
// Per-layer kernel:  Y[128,64] = W0@h[t-d] + W1@h[t] + b, then gating.
// Block = (batch, 64-column time chunk), 128 threads = 4 waves.
// Wave w owns N-tile w (16 time cols) and all 8 M-tiles -> all 4 gates for
// the same (ch, t) are in this wave's registers -> in-register gating.
// fp32 WMMA (16x16x4), K = 2 taps * 32 ch = 16 k-steps per tile.
// ---------------------------------------------------------------------------
__global__ __launch_bounds__(128)
void wave_layer_kernel(const float* __restrict__ hprev,
                       float* __restrict__ hout,
                       float* __restrict__ cbuf,
                       const float* __restrict__ convW,   // [128][32][2] this layer
                       const float* __restrict__ convB,   // [128]
                       int dil) {
  // row stride 66 (=64+2): consecutive m -> bank+2, keeps 8B alignment of pairs
  __shared__ float Wlds[128 * 66];
  // row stride 80: lane-halves (rows k and k+2) land in disjoint bank groups
  __shared__ float Hlds[64 * 80];        // [tap*32 + k][80]
  __shared__ float BiasLds[128];

  const int tid   = threadIdx.x;
  const int bb    = blockIdx.y;
  const int tbase = blockIdx.x * TN_;

  // Stage weights: Wlds[m][tap*32+k] = convW[m][k][tap]
  for (int e = tid; e < 128 * 64; e += 128) {
    int m = e >> 6, q = e & 63;
    int tap = q >> 5, k = q & 31;
    Wlds[m * 66 + q] = convW[((m * 32 + k) << 1) + tap];
  }
  BiasLds[tid] = convB[tid];

  // Stage h windows: tap0 -> cols [tbase-d, tbase-d+63], tap1 -> [tbase, tbase+63]
  for (int e = tid; e < 64 * 64; e += 128) {
    int tk = e >> 6, col = e & 63;
    int tap = tk >> 5, k = tk & 31;
    int t = tbase + col - (tap == 0 ? dil : 0);
    float v = 0.0f;                                   // causal zero padding
    if (t >= 0) v = hprev[((size_t)bb * RC_ + k) * T_ + t];
    Hlds[tk * 80 + col] = v;
  }
  __syncthreads();

  const int wave = tid >> 5;
  const int lane = tid & 31;
  const int hi   = lane >> 4;      // lane-half
  const int ln   = lane & 15;

  // Accumulators: acc[mt] element e holds Y[m = 16*mt + e + 8*hi][n = ln].
  v8f acc[8];
#pragma unroll
  for (int mt = 0; mt < 8; ++mt)
#pragma unroll
    for (int e = 0; e < 8; ++e)
      acc[mt][e] = BiasLds[16 * mt + 8 * hi + e];     // bias folded into C

  const int col = wave * 16 + ln;                     // this wave's time column
#pragma unroll
  for (int tap = 0; tap < 2; ++tap) {
#pragma unroll
    for (int kb = 0; kb < 32; kb += 4) {
      // B fragment (4x16): b[v] = H[kb + 2*hi + v][col]
      const int k0 = tap * 32 + kb + 2 * hi;
      v2f bf;
      bf.x = Hlds[k0 * 80 + col];
      bf.y = Hlds[(k0 + 1) * 80 + col];
#pragma unroll
      for (int mt = 0; mt < 8; ++mt) {
        // A fragment (16x4): a[v] = W[16*mt + ln][tap*32 + kb + 2*hi + v]
        const int ao = (16 * mt + ln) * 66 + tap * 32 + kb + 2 * hi;
        v2f af;
        af.x = Wlds[ao];
        af.y = Wlds[ao + 1];
        acc[mt] = __builtin_amdgcn_wmma_f32_16x16x4_f32(
            false, af, false, bf, (short)0, acc[mt], false, false);
      }
    }
  }

  // Gating. M-tiles [0,1]=input_gate, [2,3]=conv_filter, [4,5]=conv_gate,
  // [6,7]=emit_gate; channel ch = 16*j + 8*hi + e for tile pair offset j.
  const int t = tbase + wave * 16 + ln;
#pragma unroll
  for (int j = 0; j < 2; ++j)
#pragma unroll
    for (int e = 0; e < 8; ++e) {
      int ch = 16 * j + 8 * hi + e;
      float ig = acc[0 + j][e];
      float cf = acc[2 + j][e];
      float cg = acc[4 + j][e];
      float eg = acc[6 + j][e];
      size_t cidx = ((size_t)bb * RC_ + ch) * T_ + t;
      float co = cbuf[cidx];
      float cn = fsigmoid(ig) * co + ftanh_(cf) * fsigmoid(cg);
      cbuf[cidx] = cn;
      hout[cidx] = fsigmoid(eg) * ftanh_(cn);
    }
}

// ---------------------------------------------------------------------------
// Launch: 1 init kernel + 23 sequential layer kernels (stream-ordered).
// d_out = [24][B][RC][T] fp32; layer i reads slice i, writes slice i+1.
// d_ws holds the running cell state c ([B][RC][T] fp32 = 33.5 MB).
// ---------------------------------------------------------------------------
extern "C" void kernel_launch(void* const* d_in, const int* in_sizes, int n_in,
                              void* d_out, int out_size, void* d_ws, size_t ws_size,
                              hipStream_t stream) {
  const float* x     = (const float*)d_in[0];
  const float* feat  = (const float*)d_in[1];
  const float* w_h   = (const float*)d_in[2];
  const float* b_h   = (const float*)d_in[3];
  const float* w_c   = (const float*)d_in[4];
  const float* b_c   = (const float*)d_in[5];
  const float* convW = (const float*)d_in[6];
  const float* convB = (const float*)d_in[7];
  float* out  = (float*)d_out;
  float* cbuf = (float*)d_ws;

  const size_t S = (size_t)B_ * RC_ * T_;   // one state slice

  wave_init_kernel<<<(B_ * T_) / 256, 256, 0, stream>>>(
      x, feat, w_h, b_h, w_c, b_c, out /* slice 0 */, cbuf);

  for (int i = 0; i < NL_; ++i) {
    int dil = 1 << (i & 7);                 // [1..128] repeated
    dim3 grid(T_ / TN_, B_);
    wave_layer_kernel<<<grid, 128, 0, stream>>>(
        out + (size_t)i * S,                // h_in  = states[i]
        out + (size_t)(i + 1) * S,          // h_out = states[i+1]
        cbuf,
        convW + (size_t)i * 128 * 32 * 2,
        convB + (size_t)i * 128,
        dil);
  }
}